// ClipCluLoss_62062277427238
// MI455X (gfx1250) — compile-verified
//
#include <hip/hip_runtime.h>
#include <math.h>

typedef float v2f  __attribute__((ext_vector_type(2)));
typedef float v8f  __attribute__((ext_vector_type(8)));
typedef unsigned int u32x4 __attribute__((ext_vector_type(4)));
typedef int  i32x8 __attribute__((ext_vector_type(8)));
typedef int  i32x4 __attribute__((ext_vector_type(4)));

#define T_DIM 32
#define D_DIM 1024
// TDM inserts 1 pad DWORD every 256 DWORDs (pad_interval=7 -> 256dw, pad_amount=0 -> 1dw).
// Row of 1024 floats -> 4 pads -> effective stride 1028 floats. 1028 % 64 == 4, so the 16
// rows touched by one WMMA operand fetch land in 16 distinct LDS banks (conflict-free).
#define ROW_STRIDE 1028
// In-row address map with TDM padding: idx(col) = col + (col >> 8)

__global__ __launch_bounds__(256) void clip_gram_kernel(const float* __restrict__ x,
                                                        float* __restrict__ dist,
                                                        int B) {
  __shared__ float Xs[T_DIM * ROW_STRIDE];   // 131,584 B padded 32x1024 fp32 tile
  __shared__ float G0[T_DIM * T_DIM];        // 4 KB raw Gram
  __shared__ float invs[T_DIM];

  const int tid  = threadIdx.x;
  const int b    = blockIdx.x;
  const int wave = tid >> 5;
  const int lane = tid & 31;
  const int half = lane >> 4;   // 0: lanes 0-15, 1: lanes 16-31
  const int l15  = lane & 15;

  // Zero the Gram accumulator (covered by the barrier below).
  for (int i = tid; i < T_DIM * T_DIM; i += 256) G0[i] = 0.0f;

  // Wave 0 issues one TDM descriptor: DMA this sample's 32x1024 fp32 tile -> LDS.
  if (tid < 32) {
    unsigned long long gaddr =
        (unsigned long long)(const void*)x +
        (unsigned long long)b * (unsigned long long)(T_DIM * D_DIM * 4);
    unsigned ldsOff = (unsigned)(unsigned long long)(const void*)&Xs[0]; // low 32b of flat = LDS byte offset

    u32x4 g0;
    g0.x = 1u;                                        // count=1 (valid), user mode, no gather
    g0.y = ldsOff;                                    // lds_addr (bytes)
    g0.z = (unsigned)(gaddr & 0xFFFFFFFFull);         // global_addr[31:0]
    g0.w = ((unsigned)(gaddr >> 32) & 0x01FFFFFFu)    // global_addr[56:32]
           | (2u << 30);                              // type=2 (image)

    i32x8 g1;
    g1[0] = (2 << 16)        // data_size = 4B
          | (1 << 20)        // pad_enable
          | (7 << 22);       // pad_interval = 256 DWORDs (pad_amount field = 0 -> 1 DWORD)
    g1[1] = (int)(1024u << 16);  // tensor_dim0 = 1024 (low16 @ bits 63:48)
    g1[2] = (int)(32u << 16);    // tensor_dim1 = 32   (low16 @ bits 95:80)
    g1[3] = (int)(1024u << 16);  // tile_dim0 = 1024   (bits 127:112)
    g1[4] = 32;                  // tile_dim1 = 32 (tile_dim2 = 0)
    g1[5] = 1024;                // tensor_dim0_stride = 1024 (low32)
    g1[6] = 0;                   // stride hi16 / tensor_dim1_stride lo16
    g1[7] = 0;

    i32x4 gz4 = {0, 0, 0, 0};    // groups 2/3 unused (2D tensor)
    i32x8 gz8 = {0, 0, 0, 0, 0, 0, 0, 0};

    __builtin_amdgcn_tensor_load_to_lds(g0, g1, gz4, gz4, gz8, 0);
    __builtin_amdgcn_s_wait_tensorcnt(0);
  }
  __syncthreads();

  // ---- Gram via V_WMMA_F32_16X16X4_F32 ------------------------------------
  // A tile (16x4, rows t0..t0+15): lane(l15,half) holds K = k+2*half, k+2*half+1 (VGPR0/1).
  // B tile (4x16 = X^T chunk, cols t1..t1+15): lane(l15,half) holds rows K = k+half, k+half+2.
  const int base0  = l15 * ROW_STRIDE;          // rows 0..15
  const int base16 = (l15 + 16) * ROW_STRIDE;   // rows 16..31

  v8f c00 = {}, c01 = {}, c10 = {}, c11 = {};

  const int k0 = wave * (D_DIM / 8);            // 128-wide K-chunk per wave
#pragma unroll 4
  for (int kk = 0; kk < D_DIM / 8; kk += 4) {
    const int k  = k0 + kk;
    const int ko = k + (k >> 8);                // TDM-padded in-row offset (uniform in step)

    v2f a0, a1, b0, b1;
    a0.x = Xs[base0  + ko + 2 * half];
    a0.y = Xs[base0  + ko + 2 * half + 1];
    a1.x = Xs[base16 + ko + 2 * half];
    a1.y = Xs[base16 + ko + 2 * half + 1];
    b0.x = Xs[base0  + ko + half];
    b0.y = Xs[base0  + ko + half + 2];
    b1.x = Xs[base16 + ko + half];
    b1.y = Xs[base16 + ko + half + 2];

    c00 = __builtin_amdgcn_wmma_f32_16x16x4_f32(false, a0, false, b0, (short)0, c00, false, false);
    c01 = __builtin_amdgcn_wmma_f32_16x16x4_f32(false, a0, false, b1, (short)0, c01, false, false);
    c10 = __builtin_amdgcn_wmma_f32_16x16x4_f32(false, a1, false, b0, (short)0, c10, false, false);
    c11 = __builtin_amdgcn_wmma_f32_16x16x4_f32(false, a1, false, b1, (short)0, c11, false, false);
  }

  // Merge per-wave partial Grams into LDS G0 (ds_add_f32 atomics).
  // C layout: VGPR j -> M = j + 8*half; N = l15.
#pragma unroll
  for (int j = 0; j < 8; ++j) {
    const int m = j + 8 * half;
    const int n = l15;
    atomicAdd(&G0[m * 32 + n],               c00[j]);
    atomicAdd(&G0[m * 32 + (n + 16)],        c01[j]);
    atomicAdd(&G0[(m + 16) * 32 + n],        c10[j]);
    atomicAdd(&G0[(m + 16) * 32 + (n + 16)], c11[j]);
  }
  __syncthreads();

  // ---- Epilogue: norms from the diagonal, weighted Gram sum ---------------
  if (tid < 32) {
    const float diag = G0[tid * 32 + tid];
    invs[tid] = 1.0f / fmaxf(sqrtf(diag), 1e-12f);
  }
  __syncthreads();

  if (tid < 32) {
    const float wt = invs[tid];
    float rs = 0.0f;
#pragma unroll
    for (int n = 0; n < 32; ++n) rs += G0[tid * 32 + n] * invs[n];
    rs *= wt;
    // wave32 reduction
#pragma unroll
    for (int off = 16; off > 0; off >>= 1) rs += __shfl_down(rs, off, 32);
    if (tid == 0) {
      // dist_b = T - (1/T) * sum_{t,t'} G0 * inv_t * inv_t'
      dist[b] = (float)T_DIM - rs * (1.0f / (float)T_DIM);
    }
  }
}

__global__ __launch_bounds__(256) void final_reduce_kernel(const float* __restrict__ ws,
                                                           float* __restrict__ out,
                                                           int B) {
  __shared__ float s[256];
  float acc = 0.0f;
  for (int i = threadIdx.x; i < B; i += 256) acc += ws[i];
  s[threadIdx.x] = acc;
  __syncthreads();
  for (int off = 128; off > 0; off >>= 1) {
    if ((int)threadIdx.x < off) s[threadIdx.x] += s[threadIdx.x + off];
    __syncthreads();
  }
  if (threadIdx.x == 0) out[0] = s[0] / (float)B;
}

extern "C" void kernel_launch(void* const* d_in, const int* in_sizes, int n_in,
                              void* d_out, int out_size, void* d_ws, size_t ws_size,
                              hipStream_t stream) {
  (void)n_in; (void)out_size; (void)ws_size;
  const float* x = (const float*)d_in[0];
  float* out = (float*)d_out;
  float* ws  = (float*)d_ws;
  const int B = in_sizes[0] / (T_DIM * D_DIM);   // 1024

  clip_gram_kernel<<<B, 256, 0, stream>>>(x, ws, B);
  final_reduce_kernel<<<1, 256, 0, stream>>>(ws, out, B);
}